// AtomUpdateBlock_34797825032828
// MI455X (gfx1250) — compile-verified
//
#include <hip/hip_runtime.h>
#include <hip/hip_bf16.h>

// ---------------------------------------------------------------------------
// AtomUpdateBlock for MI455X (gfx1250, wave32, WMMA).
//  Phase 1 (memory-bound, ~75us floor @23.3TB/s for 1.74GB of m+rbf):
//    x2[a,c] = sum_{e: id_j[e]=a} m[e,c] * (rbf[e,:] @ W_rbf[:,c])
//    V_WMMA_F32_16X16X4_F32 for the K=16 GEMM (W_rbf resident in VGPRs),
//    batched immediate-offset m loads, native f32 global atomics into the
//    L2-resident 51MB accumulator.
//  Phase 2: fused silu(x2@W1) + 2 residual blocks per 64-row tile; LDS
//    ping-pong activations, double-buffered weight slabs, f32 WMMA.
// ---------------------------------------------------------------------------

typedef float v2f __attribute__((ext_vector_type(2)));
typedef float v8f __attribute__((ext_vector_type(8)));

#define EMB       256
#define EMB_RBF   16
#define INV_SQRT2 0.70710678118654752f

__device__ __forceinline__ v8f wmma_f32(v2f a, v2f b, v8f c) {
  // (neg_a, A, neg_b, B, c_mod, C, reuse_a, reuse_b)
  return __builtin_amdgcn_wmma_f32_16x16x4_f32(false, a, false, b, (short)0, c,
                                               false, false);
}

__device__ __forceinline__ float silu_f(float x) {
  return x / (1.0f + __expf(-x));
}

// ---------------------------------------------------------------------------
// Zero-init workspace (x2 accumulator).
// ---------------------------------------------------------------------------
__global__ __launch_bounds__(256) void zero_f4_kernel(float4* __restrict__ p,
                                                      int n4) {
  int i = blockIdx.x * blockDim.x + threadIdx.x;
  if (i < n4) p[i] = make_float4(0.f, 0.f, 0.f, 0.f);
}

// ---------------------------------------------------------------------------
// Edge kernel: per 16-edge tile, mlp = rbf_tile(16x16) @ W_rbf(16x256) via
// 4 WMMA k-steps; gate by m and atomically scatter into x2[id_j[e], :].
// Block = 256 threads = 8 waves.  Wave w: edge-subtile (w/4), col group
// 64*(w%4).  W_rbf B-fragments are loop-invariant -> loaded once to VGPRs.
// All per-tile memory ops hang off shared base addresses with compile-time
// immediate offsets to keep VALU off the streaming path.
// ---------------------------------------------------------------------------
__global__ __launch_bounds__(256) void edge_scatter_kernel(
    const float* __restrict__ m, const float* __restrict__ rbf,
    const int* __restrict__ id_j, const float* __restrict__ W_rbf,
    float* __restrict__ x2, int n_edges) {
  const int lane  = threadIdx.x & 31;
  const int wid   = threadIdx.x >> 5;
  const int cgrp  = wid & 3;   // 64-column group
  const int tsel  = wid >> 2;  // which 16-edge tile of the pair
  const int nlo   = lane & 15;
  const int h     = lane >> 4;
  const int ccol0 = cgrp * 64 + nlo;  // this lane's base column

  // Preload all B fragments of W_rbf: 4 k-steps x 4 col-chunks, 2 f32 each.
  v2f bfrag[4][4];
#pragma unroll
  for (int ks = 0; ks < 4; ++ks) {
    const int kb = ks * 4 + h * 2;
#pragma unroll
    for (int ct = 0; ct < 4; ++ct) {
      v2f b;
      b.x = W_rbf[kb * EMB + ccol0 + ct * 16];
      b.y = W_rbf[(kb + 1) * EMB + ccol0 + ct * 16];
      bfrag[ks][ct] = b;
    }
  }

  const int npairs = n_edges / 32;  // 32 edges per block iteration
  for (int p = blockIdx.x; p < npairs; p += gridDim.x) {
    const int e0 = p * 32 + tsel * 16;

    // Shared per-tile base pointers; everything below uses immediate offsets.
    const float* rbf_p = rbf + (size_t)(e0 + nlo) * EMB_RBF + 2 * h;
    const float* m_p   = m + (size_t)e0 * EMB + h * (8 * EMB) + ccol0;
    const int*   id_p  = id_j + e0 + 8 * h;

    v8f acc[4];
#pragma unroll
    for (int ct = 0; ct < 4; ++ct) acc[ct] = (v8f){0, 0, 0, 0, 0, 0, 0, 0};

#pragma unroll
    for (int ks = 0; ks < 4; ++ks) {
      const float2 t = *(const float2*)(rbf_p + ks * 4);
      v2f a;
      a.x = t.x;
      a.y = t.y;
#pragma unroll
      for (int ct = 0; ct < 4; ++ct) acc[ct] = wmma_f32(a, bfrag[ks][ct], acc[ct]);
    }

    // Batch all loads first (clause-friendly), then gate + scatter.
    int ids[8];
#pragma unroll
    for (int r = 0; r < 8; ++r) ids[r] = id_p[r];

    float mv[8][4];
#pragma unroll
    for (int r = 0; r < 8; ++r)
#pragma unroll
      for (int ct = 0; ct < 4; ++ct) mv[r][ct] = m_p[r * EMB + ct * 16];

#pragma unroll
    for (int r = 0; r < 8; ++r) {
      float* xp = x2 + (unsigned)ids[r] * (unsigned)EMB + (unsigned)ccol0;
#pragma unroll
      for (int ct = 0; ct < 4; ++ct) {
        __hip_atomic_fetch_add(xp + ct * 16, acc[ct][r] * mv[r][ct],
                               __ATOMIC_RELAXED, __HIP_MEMORY_SCOPE_AGENT);
      }
    }
  }
}

// ---------------------------------------------------------------------------
// Fused atom-side GEMM chain.  Block = 256 threads = 8 waves, 64-row tile.
// Waves: 2 super-rows (32 rows) x 4 col groups (64 cols); each wave owns
// 2x4 = 8 WMMA accumulators of 16x16.  Weight slabs (32x256) are double-
// buffered in LDS: slab s+1 is fetched to registers while slab s computes,
// giving one barrier per slab and full global-load overlap.
// ---------------------------------------------------------------------------
#define TM     64
#define LDSTR  264                 // padded LDS row stride in floats
#define WSLAB  (32 * LDSTR)        // one weight slab in floats

__device__ __forceinline__ void stage_slab(const float* __restrict__ Wg,
                                           int k0, float* wsbuf) {
  const int tid = threadIdx.x;
#pragma unroll
  for (int i = 0; i < 8; ++i) {
    const int p = i * 1024 + tid * 4;
    const int r = p >> 8, c = p & 255;
    *(float4*)&wsbuf[r * LDSTR + c] =
        *(const float4*)&Wg[(size_t)(k0 + r) * EMB + c];
  }
}

__device__ __forceinline__ void gemm_tile(const float* __restrict__ Wg,
                                          float* src, float* dst, float* ws,
                                          bool residual) {
  const int tid  = threadIdx.x;
  const int lane = tid & 31;
  const int wid  = tid >> 5;
  const int rb   = (wid >> 2) * 32;  // row base of this wave (2 x 16 rows)
  const int cb   = (wid & 3) * 64;   // col base (4 x 16 cols)
  const int nlo  = lane & 15;
  const int h    = lane >> 4;

  v8f acc[2][4];
#pragma unroll
  for (int rt = 0; rt < 2; ++rt)
#pragma unroll
    for (int ct = 0; ct < 4; ++ct) acc[rt][ct] = (v8f){0, 0, 0, 0, 0, 0, 0, 0};

  // Prologue: stage slab 0 into buffer 0.
  stage_slab(Wg, 0, ws);

  for (int s = 0; s < 8; ++s) {
    __syncthreads();  // ws[s&1] fully staged; prior reads of it retired
    float* wcur = ws + (s & 1) * WSLAB;
    float* wnxt = ws + ((s + 1) & 1) * WSLAB;

    // Kick off next slab's global loads into registers.
    float4 wreg[8];
    if (s < 7) {
      const int k0n = (s + 1) * 32;
      if (s < 6) __builtin_prefetch(&Wg[(size_t)(s + 2) * 32 * EMB], 0, 1);
#pragma unroll
      for (int i = 0; i < 8; ++i) {
        const int p = i * 1024 + tid * 4;
        const int r = p >> 8, c = p & 255;
        wreg[i] = *(const float4*)&Wg[(size_t)(k0n + r) * EMB + c];
      }
    }

    // Compute on current slab.
    const int k0 = s * 32;
#pragma unroll
    for (int kk = 0; kk < 32; kk += 4) {
      const int ka = k0 + kk + h * 2;
      const float2 a0t = *(const float2*)&src[(rb + nlo) * LDSTR + ka];
      const float2 a1t = *(const float2*)&src[(rb + 16 + nlo) * LDSTR + ka];
      v2f a0, a1;
      a0.x = a0t.x; a0.y = a0t.y;
      a1.x = a1t.x; a1.y = a1t.y;
      const int kb = kk + h * 2;
#pragma unroll
      for (int ct = 0; ct < 4; ++ct) {
        const int n = cb + ct * 16 + nlo;
        v2f b;
        b.x = wcur[kb * LDSTR + n];
        b.y = wcur[(kb + 1) * LDSTR + n];
        acc[0][ct] = wmma_f32(a0, b, acc[0][ct]);
        acc[1][ct] = wmma_f32(a1, b, acc[1][ct]);
      }
    }

    // Store next slab into the other buffer (safe: nobody reads it until the
    // next loop-top barrier, and all reads of it from slab s-1 retired at the
    // barrier we already passed this iteration).
    if (s < 7) {
#pragma unroll
      for (int i = 0; i < 8; ++i) {
        const int p = i * 1024 + tid * 4;
        const int r = p >> 8, c = p & 255;
        *(float4*)&wnxt[r * LDSTR + c] = wreg[i];
      }
    }
  }

  // Epilogue: silu (+ optional residual combine), write to dst tile in LDS.
#pragma unroll
  for (int rt = 0; rt < 2; ++rt)
#pragma unroll
    for (int ct = 0; ct < 4; ++ct)
#pragma unroll
      for (int r = 0; r < 8; ++r) {
        const int row = rb + rt * 16 + r + 8 * h;
        const int col = cb + ct * 16 + nlo;
        float v = silu_f(acc[rt][ct][r]);
        float* d = &dst[row * LDSTR + col];
        if (residual) v = (*d + v) * INV_SQRT2;
        *d = v;
      }
}

__global__ __launch_bounds__(256) void atom_chain_kernel(
    const float* __restrict__ x2, const float* __restrict__ W1,
    const float* __restrict__ W_res, float* __restrict__ out, int n_atoms) {
  __shared__ float xs[TM * LDSTR];
  __shared__ float ys[TM * LDSTR];
  __shared__ float ws[2 * WSLAB];  // double-buffered weight slabs

  const int row0 = blockIdx.x * TM;
  const int tid  = threadIdx.x;

  // Stage x2 tile -> ys (zero-fill out-of-range rows).
#pragma unroll
  for (int i = 0; i < 16; ++i) {
    const int p = i * 1024 + tid * 4;
    const int r = p >> 8, c = p & 255;
    float4 v = make_float4(0.f, 0.f, 0.f, 0.f);
    if (row0 + r < n_atoms) v = *(const float4*)&x2[(size_t)(row0 + r) * EMB + c];
    *(float4*)&ys[r * LDSTR + c] = v;
  }
  __syncthreads();

  // x = silu(x2 @ W1)
  gemm_tile(W1, ys, xs, ws, false);
  // residual block 0
  gemm_tile(W_res + 0 * 65536, xs, ys, ws, false);
  gemm_tile(W_res + 1 * 65536, ys, xs, ws, true);
  // residual block 1
  gemm_tile(W_res + 2 * 65536, xs, ys, ws, false);
  gemm_tile(W_res + 3 * 65536, ys, xs, ws, true);

  __syncthreads();
#pragma unroll
  for (int i = 0; i < 16; ++i) {
    const int p = i * 1024 + tid * 4;
    const int r = p >> 8, c = p & 255;
    if (row0 + r < n_atoms)
      *(float4*)&out[(size_t)(row0 + r) * EMB + c] = *(const float4*)&xs[r * LDSTR + c];
  }
}

// ---------------------------------------------------------------------------
// Launch
// ---------------------------------------------------------------------------
extern "C" void kernel_launch(void* const* d_in, const int* in_sizes, int n_in,
                              void* d_out, int out_size, void* d_ws,
                              size_t ws_size, hipStream_t stream) {
  // Inputs (setup_inputs order): h, m, rbf, id_j, W_rbf, W1, W_res
  const float* m     = (const float*)d_in[1];
  const float* rbf   = (const float*)d_in[2];
  const int*   id_j  = (const int*)d_in[3];
  const float* W_rbf = (const float*)d_in[4];
  const float* W1    = (const float*)d_in[5];
  const float* W_res = (const float*)d_in[6];
  float*       out   = (float*)d_out;

  const int n_atoms = in_sizes[0] / EMB;  // 50000
  const int n_edges = in_sizes[1] / EMB;  // 1600000

  float* x2 = (float*)d_ws;  // n_atoms * 256 f32 accumulator

  // 1) zero the scatter accumulator
  const int n4 = (n_atoms * EMB) / 4;
  zero_f4_kernel<<<(n4 + 255) / 256, 256, 0, stream>>>((float4*)x2, n4);

  // 2) edge phase: rbf-GEMM (WMMA) + gate + atomic scatter
  edge_scatter_kernel<<<2000, 256, 0, stream>>>(m, rbf, id_j, W_rbf, x2,
                                                n_edges);

  // 3) fused atom-side GEMM chain (WMMA), 64 rows per block
  atom_chain_kernel<<<(n_atoms + TM - 1) / TM, 256, 0, stream>>>(x2, W1, W_res,
                                                                 out, n_atoms);
}